// LineVectorizer_17824114279043
// MI455X (gfx1250) — compile-verified
//
#include <hip/hip_runtime.h>
#include <math.h>

// Problem constants (from reference)
#define HH      256
#define CIN     256
#define DLOI    128
#define HALF_   64
#define NPTS0   32
#define NPTS1   8
#define LNUM    5000
#define BNUM    2
#define SPATIAL (HH*HH)

#define G    8            // lines per bottleneck block
#define COLS (G*NPTS1)    // 64 GEMM columns per block

typedef float v2f __attribute__((ext_vector_type(2)));
typedef float v8f __attribute__((ext_vector_type(8)));

// D = A(16x4,f32) * B(4x16,f32) + C(16x16,f32)  -- V_WMMA_F32_16X16X4_F32
__device__ __forceinline__ v8f wmma_k4(v2f a, v2f b, v8f c) {
    return __builtin_amdgcn_wmma_f32_16x16x4_f32(false, a, false, b,
                                                 (short)0, c, false, false);
}

// ---------------------------------------------------------------------------
// Async global->LDS copy (GLOBAL_LOAD_ASYNC_TO_LDS_B32, ASYNCcnt) when the
// toolchain declares the gfx1250 builtins; otherwise VGPR-bounce fallback.
// Builtin signature (per hipcc diagnostic): (AS1 int*, AS3 int*, imm, cpol).
// ---------------------------------------------------------------------------
#if __has_builtin(__builtin_amdgcn_global_load_async_to_lds_b32) && \
    __has_builtin(__builtin_amdgcn_s_wait_asynccnt)
#define HAVE_ASYNC_LDS 1
__device__ __forceinline__ void async_copy_f32(const float* g, float* l) {
    __builtin_amdgcn_global_load_async_to_lds_b32(
        (__attribute__((address_space(1))) int*)g,
        (__attribute__((address_space(3))) int*)l,
        0, 0);
}
__device__ __forceinline__ void async_copy_wait() {
    __builtin_amdgcn_s_wait_asynccnt(0);
}
#else
#define HAVE_ASYNC_LDS 0
__device__ __forceinline__ void async_copy_f32(const float* g, float* l) {
    *l = *g;
}
__device__ __forceinline__ void async_copy_wait() {}
#endif

// ---------------------------------------------------------------------------
// Kernel 1: fc1  x[b][s][o] = sum_c feature[b][c][s] * w[o][c] + bias[o]
// Block: 256 thr = 8 waves. Block owns 16 spatial columns, wave owns 16 M-rows.
// ---------------------------------------------------------------------------
__global__ __launch_bounds__(256) void fc1_kernel(
    const float* __restrict__ feat, const float* __restrict__ w,
    const float* __restrict__ bias, float* __restrict__ x) {
    // B staged K-paired: element (c, n) at [(c>>1)*32 + n*2 + (c&1)]
    __shared__ float Blds[CIN * 16];
    __shared__ float Ct[16 * DLOI];        // [n][o] transpose buffer

    const int b   = blockIdx.y;
    const int s0  = blockIdx.x * 16;
    const int tid = threadIdx.x;
    const float* F = feat + (size_t)b * CIN * SPATIAL;

    const int n  = tid & 15;
    const int cb = tid >> 4;
#pragma unroll
    for (int i = 0; i < 16; ++i) {
        int c = cb + i * 16;
        async_copy_f32(&F[(size_t)c * SPATIAL + s0 + n],
                       &Blds[(c >> 1) * 32 + n * 2 + (c & 1)]);
    }
    async_copy_wait();
    __syncthreads();

    const int wave = tid >> 5, lane = tid & 31, lm = lane & 15, hi = lane >> 4;
    const int m0 = wave * 16;
    v8f acc = {};
#pragma unroll 4
    for (int kk = 0; kk < CIN / 4; ++kk) {
        int k0 = kk * 4;
        v2f a  = *(const v2f*)(w + (m0 + lm) * CIN + k0 + 2 * hi);
        v2f bb = *(const v2f*)(Blds + (k0 / 2 + hi) * 32 + lm * 2);
        acc = wmma_k4(a, bb, acc);
    }
#pragma unroll
    for (int i = 0; i < 8; ++i) {
        int o = m0 + i + 8 * hi;           // C layout: M = i + 8*half, N = lm
        Ct[lm * DLOI + o] = acc[i];
    }
    __syncthreads();
    for (int e = tid; e < 16 * DLOI; e += 256) {
        int s = e >> 7, o = e & 127;
        x[((size_t)b * SPATIAL + s0 + s) * DLOI + o] = Ct[e] + bias[o];
    }
}

// ---------------------------------------------------------------------------
// Kernel 2: line sampling (bilinear) + MaxPool1d(4) -> xp[n][c][p]
// Block per line, thread = channel (x is channel-minor -> coalesced gathers).
// ---------------------------------------------------------------------------
__global__ __launch_bounds__(128) void sample_kernel(
    const float* __restrict__ lines, const float* __restrict__ x,
    float* __restrict__ xp) {
    const int nIdx = blockIdx.x;           // 0..B*L-1
    const int b    = nIdx / LNUM;
    const int c    = threadIdx.x;          // 0..127
    const float* ln = lines + (size_t)nIdx * 4;
    const float p0x = ln[0], p0y = ln[1], p1x = ln[2], p1y = ln[3];
    const float* xb = x + (size_t)b * SPATIAL * DLOI;

    float m = 0.f;
#pragma unroll 4
    for (int t = 0; t < NPTS0; ++t) {
        float lam = (float)t * (1.0f / 31.0f);
        float il  = 1.0f - lam;
        float px  = p0x * lam + p1x * il - 0.5f;
        float py  = p0y * lam + p1y * il - 0.5f;
        float px0 = fminf(fmaxf(floorf(px), 0.f), 255.f);
        float py0 = fminf(fmaxf(floorf(py), 0.f), 255.f);
        float px1 = fminf(px0 + 1.f, 255.f);
        float py1 = fminf(py0 + 1.f, 255.f);
        int i0 = (int)px0, j0 = (int)py0, i1 = (int)px1, j1 = (int)py1;
        float w00 = (px1 - px) * (py1 - py);
        float w10 = (px - px0) * (py1 - py);
        float w01 = (px1 - px) * (py - py0);
        float w11 = (px - px0) * (py - py0);
        float v = xb[(size_t)(i0 * HH + j0) * DLOI + c] * w00
                + xb[(size_t)(i1 * HH + j0) * DLOI + c] * w10
                + xb[(size_t)(i0 * HH + j1) * DLOI + c] * w01
                + xb[(size_t)(i1 * HH + j1) * DLOI + c] * w11;
        m = ((t & 3) == 0) ? v : fmaxf(m, v);
        if ((t & 3) == 3)
            xp[(size_t)nIdx * (DLOI * NPTS1) + c * NPTS1 + (t >> 2)] = m;
    }
}

// ---------------------------------------------------------------------------
// Kernel 3: bottleneck (bn+relu, c1, c2 conv3, c3, residual) + fc2 + softmax
// Block: 256 thr = 8 waves; handles G=8 lines (64 columns, 4 col-tiles of 16,
// 2-way M-split per col-tile). All intermediates in LDS (~99 KB).
// ---------------------------------------------------------------------------
__global__ __launch_bounds__(256) void bottleneck_kernel(
    const float* __restrict__ xp,
    const float* __restrict__ bn1, const float* __restrict__ c1_w,
    const float* __restrict__ c1_b,
    const float* __restrict__ bn2, const float* __restrict__ c2_w,
    const float* __restrict__ c2_b,
    const float* __restrict__ bn3, const float* __restrict__ c3_w,
    const float* __restrict__ c3_b,
    const float* __restrict__ w_fc2, const float* __restrict__ b_fc2,
    float* __restrict__ out) {
    __shared__ float XP[DLOI * COLS];   // K-paired: [(c>>1)*128 + col*2 + (c&1)]
    __shared__ float RA[DLOI * COLS];   // r1 (paired); later Y (plain [o*COLS+col])
    __shared__ float RB[HALF_ * COLS];  // r2 (paired)
    __shared__ float RC[HALF_ * COLS];  // r3 (paired)
    __shared__ float sc1[DLOI], sh1[DLOI];
    __shared__ float sc2[HALF_], sh2[HALF_], sc3[HALF_], sh3[HALF_];
    __shared__ float red[256];
    __shared__ float lg[G * 4];

    const int tid = threadIdx.x;
    const int n0  = blockIdx.x * G;

    // Issue async xp-tile copy first so it overlaps the BN-param setup below.
    for (int e = tid; e < DLOI * COLS; e += 256) {
        int li = e >> 10, c = (e >> 3) & 127, p = e & 7;
        int col = li * NPTS1 + p;
        async_copy_f32(&xp[(size_t)(n0 + li) * (DLOI * NPTS1) + c * NPTS1 + p],
                       &XP[(c >> 1) * (2 * COLS) + col * 2 + (c & 1)]);
    }

    // Fold eval-mode BN into scale/shift.
    if (tid < 128) {
        int c = tid;
        float s = bn1[c] * rsqrtf(bn1[3 * 128 + c] + 1e-5f);
        sc1[c] = s; sh1[c] = bn1[128 + c] - bn1[2 * 128 + c] * s;
    } else if (tid < 192) {
        int c = tid - 128;
        float s = bn2[c] * rsqrtf(bn2[3 * 64 + c] + 1e-5f);
        sc2[c] = s; sh2[c] = bn2[64 + c] - bn2[2 * 64 + c] * s;
    } else {
        int c = tid - 192;
        float s = bn3[c] * rsqrtf(bn3[3 * 64 + c] + 1e-5f);
        sc3[c] = s; sh3[c] = bn3[64 + c] - bn3[2 * 64 + c] * s;
    }
    async_copy_wait();
    __syncthreads();
    // r1 = relu(bn1(xp))
    for (int e = tid; e < DLOI * COLS; e += 256) {
        int c = e >> 6, col = e & 63;
        int idx = (c >> 1) * (2 * COLS) + col * 2 + (c & 1);
        RA[idx] = fmaxf(XP[idx] * sc1[c] + sh1[c], 0.f);
    }
    __syncthreads();

    const int wave = tid >> 5, lane = tid & 31, lm = lane & 15, hi = lane >> 4;
    const int ctile = wave >> 1, ms = wave & 1;
    const int colL = ctile * 16 + lm;

    // ---- c1: [64 x 128] GEMM, then bn2+relu -> RB ----
#pragma unroll
    for (int mi = 0; mi < 2; ++mi) {
        int mt = ms * 2 + mi;
        v8f acc = {};
        for (int kk = 0; kk < DLOI / 4; ++kk) {
            int k0 = kk * 4;
            v2f a  = *(const v2f*)(c1_w + (mt * 16 + lm) * DLOI + k0 + 2 * hi);
            v2f bb = *(const v2f*)(RA + (k0 / 2 + hi) * (2 * COLS) + colL * 2);
            acc = wmma_k4(a, bb, acc);
        }
#pragma unroll
        for (int i = 0; i < 8; ++i) {
            int o = mt * 16 + i + 8 * hi;
            float v = (acc[i] + c1_b[o]) * sc2[o] + sh2[o];
            RB[(o >> 1) * (2 * COLS) + colL * 2 + (o & 1)] = fmaxf(v, 0.f);
        }
    }
    __syncthreads();

    // ---- c2: conv1d k=3 (pad 1) as 3 shifted GEMMs, then bn3+relu -> RC ----
    const int pL = colL & 7;
#pragma unroll
    for (int mi = 0; mi < 2; ++mi) {
        int mt = ms * 2 + mi;
        v8f acc = {};
#pragma unroll
        for (int d = 0; d < 3; ++d) {
            int sp = pL + d - 1;
            float msk = (sp >= 0 && sp < 8) ? 1.f : 0.f;  // zero padding
            int scol = colL + d - 1;
            scol = scol < 0 ? 0 : (scol > COLS - 1 ? COLS - 1 : scol);
            for (int kk = 0; kk < HALF_ / 4; ++kk) {
                int k0 = kk * 4;
                v2f a;
                a.x = c2_w[((mt * 16 + lm) * HALF_ + (k0 + 2 * hi)) * 3 + d];
                a.y = c2_w[((mt * 16 + lm) * HALF_ + (k0 + 2 * hi + 1)) * 3 + d];
                v2f bb = *(const v2f*)(RB + (k0 / 2 + hi) * (2 * COLS) + scol * 2);
                bb.x *= msk; bb.y *= msk;
                acc = wmma_k4(a, bb, acc);
            }
        }
#pragma unroll
        for (int i = 0; i < 8; ++i) {
            int o = mt * 16 + i + 8 * hi;
            float v = (acc[i] + c2_b[o]) * sc3[o] + sh3[o];
            RC[(o >> 1) * (2 * COLS) + colL * 2 + (o & 1)] = fmaxf(v, 0.f);
        }
    }
    __syncthreads();

    // ---- c3: [128 x 64] GEMM + residual; Y (plain layout) -> RA ----
#pragma unroll
    for (int mi = 0; mi < 4; ++mi) {
        int mt = ms * 4 + mi;
        v8f acc = {};
        for (int kk = 0; kk < HALF_ / 4; ++kk) {
            int k0 = kk * 4;
            v2f a  = *(const v2f*)(c3_w + (mt * 16 + lm) * HALF_ + k0 + 2 * hi);
            v2f bb = *(const v2f*)(RC + (k0 / 2 + hi) * (2 * COLS) + colL * 2);
            acc = wmma_k4(a, bb, acc);
        }
#pragma unroll
        for (int i = 0; i < 8; ++i) {
            int o = mt * 16 + i + 8 * hi;
            float y = XP[(o >> 1) * (2 * COLS) + colL * 2 + (o & 1)]
                    + acc[i] + c3_b[o];
            RA[o * COLS + colL] = y;                 // plain [o][col]
        }
    }
    __syncthreads();

    // ---- fc2: relu(flat) @ W^T (+b), softmax ----
    {
        int li = tid >> 5;            // line 0..7
        int q  = (tid >> 3) & 3;      // logit 0..3
        int r  = tid & 7;             // 8-way K split
        float s = 0.f;
        for (int oi = 0; oi < 16; ++oi) {
            int o = r + oi * 8;
            const float* wq = w_fc2 + (size_t)q * (DLOI * NPTS1) + o * NPTS1;
            const float* yy = RA + o * COLS + li * NPTS1;
#pragma unroll
            for (int p = 0; p < 8; ++p)
                s += fmaxf(yy[p], 0.f) * wq[p];
        }
        red[tid] = s;
    }
    __syncthreads();
    if ((tid & 7) == 0) {
        float s = 0.f;
#pragma unroll
        for (int k = 0; k < 8; ++k) s += red[tid + k];
        int li = tid >> 5, q = (tid >> 3) & 3;
        lg[li * 4 + q] = s + b_fc2[q];
    }
    __syncthreads();
    if (tid < G) {
        int li = tid;
        float l0 = lg[li * 4 + 0], l1 = lg[li * 4 + 1];
        float l2 = lg[li * 4 + 2], l3 = lg[li * 4 + 3];
        float mx = fmaxf(fmaxf(l0, l1), fmaxf(l2, l3));
        float e0 = __expf(l0 - mx), e1 = __expf(l1 - mx);
        float e2 = __expf(l2 - mx), e3 = __expf(l3 - mx);
        float inv = 1.f / (e0 + e1 + e2 + e3);
        size_t base = (size_t)(n0 + li) * 4;
        out[base + 0] = e0 * inv; out[base + 1] = e1 * inv;
        out[base + 2] = e2 * inv; out[base + 3] = e3 * inv;
    }
}

// ---------------------------------------------------------------------------
extern "C" void kernel_launch(void* const* d_in, const int* in_sizes, int n_in,
                              void* d_out, int out_size, void* d_ws, size_t ws_size,
                              hipStream_t stream) {
    (void)in_sizes; (void)n_in; (void)out_size; (void)ws_size;
    const float* feature = (const float*)d_in[0];
    const float* lines   = (const float*)d_in[1];
    const float* w_fc1   = (const float*)d_in[2];
    const float* b_fc1   = (const float*)d_in[3];
    const float* bn1     = (const float*)d_in[4];
    const float* c1_w    = (const float*)d_in[5];
    const float* c1_b    = (const float*)d_in[6];
    const float* bn2     = (const float*)d_in[7];
    const float* c2_w    = (const float*)d_in[8];
    const float* c2_b    = (const float*)d_in[9];
    const float* bn3     = (const float*)d_in[10];
    const float* c3_w    = (const float*)d_in[11];
    const float* c3_b    = (const float*)d_in[12];
    const float* w_fc2   = (const float*)d_in[13];
    const float* b_fc2   = (const float*)d_in[14];
    float* out = (float*)d_out;

    // Workspace: x [2][65536][128] f32 (64 MB) + xp [10000][128][8] f32 (41 MB)
    float* x  = (float*)d_ws;
    float* xp = x + (size_t)BNUM * SPATIAL * DLOI;

    fc1_kernel<<<dim3(SPATIAL / 16, BNUM), 256, 0, stream>>>(feature, w_fc1, b_fc1, x);
    sample_kernel<<<dim3(BNUM * LNUM), 128, 0, stream>>>(lines, x, xp);
    bottleneck_kernel<<<dim3(BNUM * LNUM / G), 256, 0, stream>>>(
        xp, bn1, c1_w, c1_b, bn2, c2_w, c2_b, bn3, c3_w, c3_b, w_fc2, b_fc2, out);
}